// GraphNet_3521873183574
// MI455X (gfx1250) — compile-verified
//
#include <hip/hip_runtime.h>
#include <hip/hip_bf16.h>

typedef __attribute__((ext_vector_type(2))) float v2f;
typedef __attribute__((ext_vector_type(8))) float v8f;

#define SLOPE 0.2f

// ---------- helpers ----------
__device__ __forceinline__ unsigned f2o(float f) {
    unsigned u = __float_as_uint(f);
    return (u & 0x80000000u) ? ~u : (u | 0x80000000u);
}
__device__ __forceinline__ float o2f(unsigned u) {
    return (u & 0x80000000u) ? __uint_as_float(u & 0x7fffffffu)
                             : __uint_as_float(~u);
}

__global__ void fill_f32(float* __restrict__ p, float v, size_t n) {
    size_t i = (size_t)blockIdx.x * blockDim.x + threadIdx.x;
    if (i < n) p[i] = v;
}

// ---------- gcn_norm ----------
__global__ void deg_kernel(const int* __restrict__ row, float* __restrict__ deg, int E) {
    int e = blockIdx.x * blockDim.x + threadIdx.x;
    if (e < E) atomicAdd(&deg[row[e]], 1.0f);
}
__global__ void dinv_kernel(float* __restrict__ deg, int N) {
    int i = blockIdx.x * blockDim.x + threadIdx.x;
    if (i < N) {
        float d = deg[i];
        deg[i] = d > 0.0f ? rsqrtf(d) : 0.0f;
    }
}
__global__ void ew_kernel(const int* __restrict__ row, const int* __restrict__ col,
                          const float* __restrict__ dinv, float* __restrict__ ew,
                          int E, int N) {
    int e = blockIdx.x * blockDim.x + threadIdx.x;
    if (e >= E + N) return;
    int r = (e < E) ? row[e] : (e - E);
    int c = (e < E) ? col[e] : (e - E);
    ew[e] = dinv[r] * dinv[c];
}

// ---------- GEMM1: xt[N,128] = x[N,256] @ W1[256,128] (f32 WMMA) ----------
// Block = 256 threads (8 waves). LDS-stage a 16x256 x-tile; wave w computes
// the 16x16 output tile at columns [16w,16w+16) via 64 v_wmma_f32_16x16x4_f32.
__global__ __launch_bounds__(256) void gemm1_wmma(const float* __restrict__ x,
                                                  const float* __restrict__ W,
                                                  float* __restrict__ xt, int N) {
    __shared__ float xs[16][260];  // stride 260 -> lanes hit distinct LDS banks
    const int tid = threadIdx.x;
    const int rowBase = blockIdx.x * 16;
    for (int i = tid; i < 16 * 64; i += 256) {
        int r = i >> 6;
        int c4 = (i & 63) << 2;
        int rr = min(rowBase + r, N - 1);
        const float4 v = *(const float4*)(x + (size_t)rr * 256 + c4);
        xs[r][c4 + 0] = v.x; xs[r][c4 + 1] = v.y;
        xs[r][c4 + 2] = v.z; xs[r][c4 + 3] = v.w;
    }
    __syncthreads();

    const int wave = tid >> 5;
    const int lane = tid & 31;
    const int m = lane & 15;       // M (for A) / N (for B) index
    const int half = lane >> 4;    // lane half selects K pair
    const int colBase = wave * 16;

    v8f acc = {};
    for (int k = 0; k < 256; k += 4) {
        v2f a, b;
        // A 16x4 (MxK): VGPR v, half h -> K = k + 2h + v, M = lane%16
        a.x = xs[m][k + 2 * half + 0];
        a.y = xs[m][k + 2 * half + 1];
        // B 4x16 (KxN): VGPR v, half h -> K = k + 2h + v, N = lane%16
        b.x = W[(size_t)(k + 2 * half + 0) * 128 + colBase + m];
        b.y = W[(size_t)(k + 2 * half + 1) * 128 + colBase + m];
        acc = __builtin_amdgcn_wmma_f32_16x16x4_f32(false, a, false, b,
                                                    (short)0, acc, false, false);
    }
    // C/D: VGPR r, half h -> M = r + 8h, N = lane%16
    #pragma unroll
    for (int r = 0; r < 8; ++r) {
        int outRow = rowBase + r + 8 * half;
        if (outRow < N) xt[(size_t)outRow * 128 + colBase + m] = acc[r];
    }
}

// ---------- GEMM2: xt2[N,16] = h[N,128] @ W2[128,16] ----------
// Each wave owns one 16-row tile; 8 waves per block.
__global__ __launch_bounds__(256) void gemm2_wmma(const float* __restrict__ h,
                                                  const float* __restrict__ W2,
                                                  float* __restrict__ xt2,
                                                  int N, int nRowTiles) {
    const int wave = threadIdx.x >> 5;
    const int lane = threadIdx.x & 31;
    const int rowTile = blockIdx.x * 8 + wave;
    if (rowTile >= nRowTiles) return;  // wave-uniform: EXEC stays all-ones
    const int rowBase = rowTile * 16;
    const int m = lane & 15;
    const int half = lane >> 4;

    const int aRow = min(rowBase + m, N - 1);
    v8f acc = {};
    for (int k = 0; k < 128; k += 4) {
        v2f a, b;
        a.x = h[(size_t)aRow * 128 + k + 2 * half + 0];
        a.y = h[(size_t)aRow * 128 + k + 2 * half + 1];
        b.x = W2[(size_t)(k + 2 * half + 0) * 16 + m];
        b.y = W2[(size_t)(k + 2 * half + 1) * 16 + m];
        acc = __builtin_amdgcn_wmma_f32_16x16x4_f32(false, a, false, b,
                                                    (short)0, acc, false, false);
    }
    #pragma unroll
    for (int r = 0; r < 8; ++r) {
        int outRow = rowBase + r + 8 * half;
        if (outRow < N) xt2[(size_t)outRow * 16 + m] = acc[r];
    }
}

// ---------- per-node attention scalars: ai = xt.wl, aj = xt.wr ----------
template <int H, int C>
__global__ void attn_scal(const float* __restrict__ xt, const float* __restrict__ att,
                          float* __restrict__ ai, float* __restrict__ aj, int N) {
    int idx = blockIdx.x * blockDim.x + threadIdx.x;  // node*H + head
    if (idx >= N * H) return;
    int node = idx / H, h = idx % H;
    const float* v = xt + (size_t)node * (H * C) + h * C;
    const float* wl = att + h * (2 * C);
    const float* wr = wl + C;
    float sl = 0.f, sr = 0.f;
    #pragma unroll
    for (int c = 0; c < C; ++c) {
        float t = v[c];
        sl = fmaf(t, wl[c], sl);
        sr = fmaf(t, wr[c], sr);
    }
    ai[idx] = sl;
    aj[idx] = sr;
}

// ---------- edge pass 1: alpha = lrelu(ai[col]+aj[row]); segment max ----------
template <int H>
__global__ void alpha_max(const int* __restrict__ row, const int* __restrict__ col,
                          const float* __restrict__ ai, const float* __restrict__ aj,
                          float* __restrict__ alpha, unsigned* __restrict__ amax,
                          int E, int N) {
    int e = blockIdx.x * blockDim.x + threadIdx.x;
    if (e >= E + N) return;
    int r = (e < E) ? row[e] : (e - E);
    int c = (e < E) ? col[e] : (e - E);
    #pragma unroll
    for (int h = 0; h < H; ++h) {
        float a = ai[c * H + h] + aj[r * H + h];
        a = (a >= 0.f) ? a : SLOPE * a;
        alpha[(size_t)e * H + h] = a;
        atomicMax(&amax[c * H + h], f2o(a));
    }
}

// ---------- edge pass 2: ex = exp(alpha - amax[col]); segment sum ----------
template <int H>
__global__ void exp_sum(const int* __restrict__ col, float* __restrict__ alpha,
                        const unsigned* __restrict__ amax, float* __restrict__ asum,
                        int E, int N) {
    int e = blockIdx.x * blockDim.x + threadIdx.x;
    if (e >= E + N) return;
    int c = (e < E) ? col[e] : (e - E);
    #pragma unroll
    for (int h = 0; h < H; ++h) {
        float ex = __expf(alpha[(size_t)e * H + h] - o2f(amax[c * H + h]));
        alpha[(size_t)e * H + h] = ex;  // store ex in place
        atomicAdd(&asum[c * H + h], ex);
    }
}

// ---------- edge pass 3 (layer1): wave per edge, lane = channel-in-head ----------
__global__ __launch_bounds__(256) void aggregate1(const int* __restrict__ row,
                                                  const int* __restrict__ col,
                                                  const float* __restrict__ ew,
                                                  const float* __restrict__ ex,
                                                  const float* __restrict__ asum,
                                                  const float* __restrict__ xt,
                                                  float* __restrict__ hout,
                                                  int E, int N) {
    int eg = blockIdx.x * 8 + (threadIdx.x >> 5);
    int lane = threadIdx.x & 31;
    if (eg >= E + N) return;
    int r = (eg < E) ? row[eg] : (eg - E);
    int c = (eg < E) ? col[eg] : (eg - E);
    float w = ew[eg];
    const float4 exv = *(const float4*)(ex + (size_t)eg * 4);
    const float4 asv = *(const float4*)(asum + (size_t)c * 4);
    float coef0 = w * exv.x / (asv.x + 1e-16f);
    float coef1 = w * exv.y / (asv.y + 1e-16f);
    float coef2 = w * exv.z / (asv.z + 1e-16f);
    float coef3 = w * exv.w / (asv.w + 1e-16f);
    const float* src = xt + (size_t)r * 128 + lane;
    float* dst = hout + (size_t)c * 128 + lane;
    atomicAdd(dst + 0,  coef0 * src[0]);
    atomicAdd(dst + 32, coef1 * src[32]);
    atomicAdd(dst + 64, coef2 * src[64]);
    atomicAdd(dst + 96, coef3 * src[96]);
}

// ---------- edge pass 3 (layer2, H=1,C=16): thread per (edge,channel) ----------
__global__ void aggregate2(const int* __restrict__ row, const int* __restrict__ col,
                           const float* __restrict__ ew, const float* __restrict__ ex,
                           const float* __restrict__ asum, const float* __restrict__ xt2,
                           float* __restrict__ out2, int E, int N) {
    long idx = (long)blockIdx.x * blockDim.x + threadIdx.x;
    if (idx >= (long)(E + N) * 16) return;
    int e = (int)(idx >> 4);
    int ch = (int)(idx & 15);
    int r = (e < E) ? row[e] : (e - E);
    int c = (e < E) ? col[e] : (e - E);
    float coef = ew[e] * ex[e] / (asum[c] + 1e-16f);
    atomicAdd(&out2[(size_t)c * 16 + ch], coef * xt2[(size_t)r * 16 + ch]);
}

// ---------- epilogues ----------
__global__ void bias_relu(float* __restrict__ h, const float* __restrict__ b, size_t n) {
    size_t i = (size_t)blockIdx.x * blockDim.x + threadIdx.x;
    if (i < n) {
        float v = h[i] + b[i & 127];
        h[i] = v > 0.f ? v : 0.f;
    }
}
__global__ void final_bias(const float* __restrict__ acc, const float* __restrict__ b,
                           float* __restrict__ out, size_t n) {
    size_t i = (size_t)blockIdx.x * blockDim.x + threadIdx.x;
    if (i < n) out[i] = acc[i] + b[i & 15];
}

static inline int cdiv(long a, long b) { return (int)((a + b - 1) / b); }

extern "C" void kernel_launch(void* const* d_in, const int* in_sizes, int n_in,
                              void* d_out, int out_size, void* d_ws, size_t ws_size,
                              hipStream_t stream) {
    const float* x    = (const float*)d_in[0];
    const int*   ei   = (const int*)d_in[1];
    const float* W1   = (const float*)d_in[2];
    const float* att1 = (const float*)d_in[3];
    const float* b1   = (const float*)d_in[4];
    const float* W2   = (const float*)d_in[5];
    const float* att2 = (const float*)d_in[6];
    const float* b2   = (const float*)d_in[7];
    float* out = (float*)d_out;

    const int N = in_sizes[0] / 256;
    const int E = in_sizes[1] / 2;
    const int EN = E + N;
    const int* row = ei;
    const int* col = ei + E;

    // workspace layout (floats)
    float* ws = (float*)d_ws;
    size_t o = 0;
    float*    deg    = ws + o; o += N;               // becomes dinv in place
    float*    ew     = ws + o; o += EN;
    float*    xt1    = ws + o; o += (size_t)N * 128;
    float*    ai1    = ws + o; o += (size_t)N * 4;
    float*    aj1    = ws + o; o += (size_t)N * 4;
    unsigned* amax1  = (unsigned*)(ws + o); o += (size_t)N * 4;
    float*    asum1  = ws + o; o += (size_t)N * 4;
    float*    alpha1 = ws + o; o += (size_t)EN * 4;  // becomes ex1 in place
    float*    hout   = ws + o; o += (size_t)N * 128; // becomes h in place
    float*    xt2    = ws + o; o += (size_t)N * 16;
    float*    ai2    = ws + o; o += N;
    float*    aj2    = ws + o; o += N;
    unsigned* amax2  = (unsigned*)(ws + o); o += N;
    float*    asum2  = ws + o; o += N;
    float*    alpha2 = ws + o; o += EN;
    float*    out2   = ws + o; o += (size_t)N * 16;

    const int B = 256;

    // init
    fill_f32<<<cdiv(N, B), B, 0, stream>>>(deg, 1.0f, (size_t)N);   // +1 self loop
    fill_f32<<<cdiv((long)N * 4, B), B, 0, stream>>>((float*)amax1, 0.0f, (size_t)N * 4);
    fill_f32<<<cdiv((long)N * 4, B), B, 0, stream>>>(asum1, 0.0f, (size_t)N * 4);
    fill_f32<<<cdiv((long)N * 128, B), B, 0, stream>>>(hout, 0.0f, (size_t)N * 128);
    fill_f32<<<cdiv(N, B), B, 0, stream>>>((float*)amax2, 0.0f, (size_t)N);
    fill_f32<<<cdiv(N, B), B, 0, stream>>>(asum2, 0.0f, (size_t)N);
    fill_f32<<<cdiv((long)N * 16, B), B, 0, stream>>>(out2, 0.0f, (size_t)N * 16);

    // gcn_norm
    deg_kernel<<<cdiv(E, B), B, 0, stream>>>(row, deg, E);
    dinv_kernel<<<cdiv(N, B), B, 0, stream>>>(deg, N);
    ew_kernel<<<cdiv(EN, B), B, 0, stream>>>(row, col, deg, ew, E, N);

    // layer 1
    gemm1_wmma<<<cdiv(N, 16), B, 0, stream>>>(x, W1, xt1, N);
    attn_scal<4, 32><<<cdiv((long)N * 4, B), B, 0, stream>>>(xt1, att1, ai1, aj1, N);
    alpha_max<4><<<cdiv(EN, B), B, 0, stream>>>(row, col, ai1, aj1, alpha1, amax1, E, N);
    exp_sum<4><<<cdiv(EN, B), B, 0, stream>>>(col, alpha1, amax1, asum1, E, N);
    aggregate1<<<cdiv(EN, 8), B, 0, stream>>>(row, col, ew, alpha1, asum1, xt1, hout, E, N);
    bias_relu<<<cdiv((long)N * 128, B), B, 0, stream>>>(hout, b1, (size_t)N * 128);

    // layer 2
    const int nRowTiles = cdiv(N, 16);
    gemm2_wmma<<<cdiv(nRowTiles, 8), B, 0, stream>>>(hout, W2, xt2, N, nRowTiles);
    attn_scal<1, 16><<<cdiv(N, B), B, 0, stream>>>(xt2, att2, ai2, aj2, N);
    alpha_max<1><<<cdiv(EN, B), B, 0, stream>>>(row, col, ai2, aj2, alpha2, amax2, E, N);
    exp_sum<1><<<cdiv(EN, B), B, 0, stream>>>(col, alpha2, amax2, asum2, E, N);
    aggregate2<<<cdiv((long)EN * 16, B), B, 0, stream>>>(row, col, ew, alpha2, asum2, xt2, out2, E, N);
    final_bias<<<cdiv((long)N * 16, B), B, 0, stream>>>(out2, b2, out, (size_t)N * 16);
}